// FusedGCN_30382598651973
// MI455X (gfx1250) — compile-verified
//
#include <hip/hip_runtime.h>

typedef __attribute__((ext_vector_type(2))) float v2f;
typedef __attribute__((ext_vector_type(8))) float v8f;

#define N_NODES 50000
#define N_EDGES 800000
#define CCH 64   // channels (IN_C == OUT_C == 64)

// ---------------------------------------------------------------------------
// Zero workspace (deg + agg) — grid-stride
// ---------------------------------------------------------------------------
__global__ void gcn_zero_kernel(float* __restrict__ p, int n) {
    int i = blockIdx.x * blockDim.x + threadIdx.x;
    int stride = gridDim.x * blockDim.x;
    for (; i < n; i += stride) p[i] = 0.0f;
}

// ---------------------------------------------------------------------------
// Stage 1: degree of target nodes (scatter-add of ones)
// ---------------------------------------------------------------------------
__global__ void gcn_degree_kernel(const long long* __restrict__ ei,
                                  float* __restrict__ deg) {
    int e = blockIdx.x * blockDim.x + threadIdx.x;
    if (e < N_EDGES) {
        int tgt = (int)ei[N_EDGES + e];
        atomicAdd(&deg[tgt], 1.0f);
    }
}

// ---------------------------------------------------------------------------
// deg -> clip(deg,1)^-0.5 in place
// ---------------------------------------------------------------------------
__global__ void gcn_rsqrt_kernel(float* __restrict__ deg) {
    int n = blockIdx.x * blockDim.x + threadIdx.x;
    if (n < N_NODES) {
        float d = deg[n];
        d = d < 1.0f ? 1.0f : d;
        deg[n] = rsqrtf(d);
    }
}

// ---------------------------------------------------------------------------
// Stages 2-4: gather x[src], scale by dis[src]*dis[tgt], scatter-add into agg.
// One wave32 per edge; each lane handles 2 channels (float2 gather, 2 f32
// atomics). All 64 channel lanes' x loads are fully coalesced (256B row).
// edge/dis loads are same-address within the wave -> single merged request.
// ---------------------------------------------------------------------------
__global__ void gcn_scatter_kernel(const float* __restrict__ x,
                                   const long long* __restrict__ ei,
                                   const float* __restrict__ dis,
                                   float* __restrict__ agg) {
    int e    = blockIdx.x * (blockDim.x >> 5) + (threadIdx.x >> 5);
    int lane = threadIdx.x & 31;
    if (e >= N_EDGES) return;
    int src = (int)ei[e];
    int tgt = (int)ei[N_EDGES + e];
    float norm = dis[src] * dis[tgt];
    const v2f xv = *(const v2f*)(x + (size_t)src * CCH + 2 * lane);
    float* a = agg + (size_t)tgt * CCH + 2 * lane;
    atomicAdd(a + 0, xv.x * norm);
    atomicAdd(a + 1, xv.y * norm);
}

// ---------------------------------------------------------------------------
// Stages 5-6: out = relu(agg @ W^T + b) via V_WMMA_F32_16X16X4_F32.
// Block = 128 threads = 4 waves; block b owns M-tile rows [16b,16b+16);
// wave w owns N-tile cols [16w,16w+16). K=64 -> 16 WMMA ops, fully unrolled.
//
// f32 16x4 A layout (ISA 7.12.2): lanes 0-15 hold K={k0,k0+1} for M=lane,
// lanes 16-31 hold K={k0+2,k0+3} for M=lane-16  -> one aligned float2/lane.
// B (4x16) mirrors it with N=lane%16, row K per half-wave.
// C/D 16x16 f32: VGPR v <-> row v + 8*(lane>=16), col lane%16.
// out[m][n] = sum_k agg[m][k] * W[n][k]  (W stored [OUT_C][IN_C] row-major).
// ---------------------------------------------------------------------------
__global__ void __launch_bounds__(128)
gcn_gemm_relu_wmma(const float* __restrict__ agg,
                   const float* __restrict__ W,
                   const float* __restrict__ bias,
                   float* __restrict__ out) {
    const int wave = threadIdx.x >> 5;   // 0..3 -> N tile
    const int lane = threadIdx.x & 31;
    const int half = lane >> 4;          // 0 | 1
    const int r    = lane & 15;
    const int m0   = blockIdx.x * 16;
    const int n0   = wave * 16;

    v8f acc = {};
#pragma unroll
    for (int k0 = 0; k0 < CCH; k0 += 4) {
        const int kk = k0 + 2 * half;
        v2f a  = *(const v2f*)(agg + (size_t)(m0 + r) * CCH + kk);
        v2f bf = *(const v2f*)(W   + (size_t)(n0 + r) * CCH + kk);
        acc = __builtin_amdgcn_wmma_f32_16x16x4_f32(
            /*neg_a=*/false, a, /*neg_b=*/false, bf,
            /*c_mod=*/(short)0, acc, /*reuse_a=*/false, /*reuse_b=*/false);
    }

    const float bv = bias[n0 + r];
#pragma unroll
    for (int v = 0; v < 8; ++v) {
        const int row = m0 + v + 8 * half;
        float val = acc[v] + bv;
        out[(size_t)row * CCH + n0 + r] = val > 0.0f ? val : 0.0f;
    }
}

// ---------------------------------------------------------------------------
// Host launch
// ---------------------------------------------------------------------------
extern "C" void kernel_launch(void* const* d_in, const int* in_sizes, int n_in,
                              void* d_out, int out_size, void* d_ws, size_t ws_size,
                              hipStream_t stream) {
    const float*     x  = (const float*)d_in[0];
    const long long* ei = (const long long*)d_in[1];   // int64 edge_index [2,E]
    const float*     W  = (const float*)d_in[2];       // [OUT_C, IN_C]
    const float*     b  = (const float*)d_in[3];       // [OUT_C]
    float*           out = (float*)d_out;              // [N, OUT_C]

    float* deg = (float*)d_ws;          // N_NODES floats (becomes deg^-1/2)
    float* agg = deg + N_NODES;         // N_NODES*CCH floats (16B-aligned)

    // zero deg + agg
    const int ztotal = N_NODES + N_NODES * CCH;
    gcn_zero_kernel<<<2048, 256, 0, stream>>>(deg, ztotal);

    // degree scatter-add
    gcn_degree_kernel<<<(N_EDGES + 255) / 256, 256, 0, stream>>>(ei, deg);

    // deg -> clip(deg,1)^-0.5
    gcn_rsqrt_kernel<<<(N_NODES + 255) / 256, 256, 0, stream>>>(deg);

    // gather/scale/scatter: one wave per edge -> 8 edges per 256-thread block
    gcn_scatter_kernel<<<(N_EDGES + 7) / 8, 256, 0, stream>>>(x, ei, deg, agg);

    // WMMA GEMM + bias + relu: 3125 M-tiles x 4 N-tile waves
    gcn_gemm_relu_wmma<<<N_NODES / 16, 128, 0, stream>>>(agg, W, b, out);
}